// GAT_4552665333904
// MI455X (gfx1250) — compile-verified
//
#include <hip/hip_runtime.h>
#include <math.h>

#define D 128
#define HD 128            // hidden dim
#define NEG_SLOPE 0.2f

#if defined(__has_builtin)
#if __has_builtin(__builtin_amdgcn_wmma_f32_16x16x4_f32)
#define HAVE_WMMA_F32 1
#endif
#endif

typedef __attribute__((ext_vector_type(2))) float v2f;
typedef __attribute__((ext_vector_type(8))) float v8f;

// ---------------------------------------------------------------------------
// Generic zero fill
__global__ void zero_kernel(float* __restrict__ p, int n) {
  int i = blockIdx.x * blockDim.x + threadIdx.x;
  if (i < n) p[i] = 0.0f;
}

// init per-node softmax state: m = -inf, denom = 0
__global__ void init_node_kernel(float* __restrict__ m, float* __restrict__ denom, int n) {
  int i = blockIdx.x * blockDim.x + threadIdx.x;
  if (i < n) { m[i] = -3.402823466e38f; denom[i] = 0.0f; }
}

// ---------------------------------------------------------------------------
// wc_src = W_src @ att_src,  wc_dst = W_dst @ att_dst   (both length-128)
// Single block of 128 threads; weights are tiny and L2/WGP$-resident.
__global__ __launch_bounds__(128) void combine_att_kernel(
    const float* __restrict__ Wsrc, const float* __restrict__ att_src,
    const float* __restrict__ Wdst, const float* __restrict__ att_dst,
    float* __restrict__ wcs, float* __restrict__ wcd) {
  int i = threadIdx.x;
  float s = 0.0f, d = 0.0f;
  for (int j = 0; j < HD; ++j) {
    s += Wsrc[i * HD + j] * att_src[j];
    d += Wdst[i * HD + j] * att_dst[j];
  }
  wcs[i] = s;
  wcd[i] = d;
}

// ---------------------------------------------------------------------------
// a_src[n] = dot(x[n,:], wc_src); a_dst[n] = dot(x[n,:], wc_dst)
// One wave32 per node, float4 per lane, shfl_xor tree reduction.
__global__ __launch_bounds__(256) void attvec_kernel(
    const float* __restrict__ X, const float* __restrict__ wcs,
    const float* __restrict__ wcd, float* __restrict__ a_s,
    float* __restrict__ a_d, int N) {
  int node = blockIdx.x * 8 + (threadIdx.x >> 5);
  int lane = threadIdx.x & 31;
  if (node >= N) return;
  const float4 xv = ((const float4*)(X + (size_t)node * D))[lane];
  const float4 sv = ((const float4*)wcs)[lane];
  const float4 dv = ((const float4*)wcd)[lane];
  float ss = xv.x * sv.x + xv.y * sv.y + xv.z * sv.z + xv.w * sv.w;
  float sd = xv.x * dv.x + xv.y * dv.y + xv.z * dv.z + xv.w * dv.w;
  for (int o = 16; o > 0; o >>= 1) {
    ss += __shfl_xor(ss, o, 32);
    sd += __shfl_xor(sd, o, 32);
  }
  if (lane == 0) { a_s[node] = ss; a_d[node] = sd; }
}

// ---------------------------------------------------------------------------
// h_src = X @ W  via V_WMMA_F32_16X16X4_F32, fp32 end to end.
// Block = 256 threads (8 waves): 16 output rows x 128 cols.
// Wave w owns the 16x16 tile at cols [16w, 16w+16). K = 128 -> 32 WMMA ops.
// x tile staged in LDS with row stride 132 floats (64-bank conflict free for
// the strided A-fragment reads).
__global__ __launch_bounds__(256) void gemm_hsrc_kernel(
    const float* __restrict__ X, const float* __restrict__ W,
    float* __restrict__ Hout, int N) {
  __shared__ float lx[16 * 132];
  const int rowBase = blockIdx.x * 16;

  for (int i = threadIdx.x; i < 16 * D; i += 256) {
    int r = i >> 7, c = i & 127;
    lx[r * 132 + c] = X[(size_t)(rowBase + r) * D + c];
  }
  __syncthreads();

  const int wave = threadIdx.x >> 5;        // 0..7 -> column tile
  const int lane = threadIdx.x & 31;
  const int colBase = wave * 16;
  const int hrow = lane & 15;               // A-fragment row (M)
  const int koff = (lane >> 4) << 1;        // 0 for lanes 0-15, 2 for 16-31
  const int col = colBase + (lane & 15);    // B/D column (N)

#ifdef HAVE_WMMA_F32
  v8f acc = {};
  for (int kk = 0; kk < D; kk += 4) {
    v2f a, b;
    // A 16x4 f32 layout: v0 = K=kk+koff, v1 = K=kk+koff+1 (ISA 7.12.2)
    a.x = lx[hrow * 132 + kk + koff];
    a.y = lx[hrow * 132 + kk + koff + 1];
    // B 4x16 f32 layout mirrors A with N across lanes
    b.x = W[(size_t)(kk + koff) * HD + col];
    b.y = W[(size_t)(kk + koff + 1) * HD + col];
    acc = __builtin_amdgcn_wmma_f32_16x16x4_f32(
        /*neg_a=*/false, a, /*neg_b=*/false, b,
        /*c_mod=*/(short)0, acc, /*reuse_a=*/false, /*reuse_b=*/false);
  }
  // D layout: VGPR r -> M = r (lanes 0-15) or r+8 (lanes 16-31), N = lane&15
  for (int r = 0; r < 8; ++r) {
    int row = rowBase + r + ((lane >> 4) << 3);
    Hout[(size_t)row * HD + col] = acc[r];
  }
#else
  // Correct scalar fallback (compiles everywhere; only used if the f32 WMMA
  // builtin is absent — histogram will show wmma==0 and we iterate).
  float accs[8] = {};
  for (int k = 0; k < D; ++k) {
    float bv = W[(size_t)k * HD + col];
    for (int r = 0; r < 8; ++r)
      accs[r] += lx[(r + ((lane >> 4) << 3)) * 132 + k] * bv;
  }
  for (int r = 0; r < 8; ++r) {
    int row = rowBase + r + ((lane >> 4) << 3);
    Hout[(size_t)row * HD + col] = accs[r];
  }
#endif
}

// ---------------------------------------------------------------------------
// Float atomic max via monotonic integer-bits trick (no CAS loop):
//   val >= 0 -> signed atomicMax on bits; val < 0 -> unsigned atomicMin.
// Requires init to -inf (0xFF800000). Lowers to GLOBAL_ATOMIC_MAX_I32 /
// GLOBAL_ATOMIC_MIN_U32.
__device__ __forceinline__ void atomicMaxF(float* addr, float val) {
  if (val >= 0.0f)
    atomicMax((int*)addr, __float_as_int(val));
  else
    atomicMin((unsigned int*)addr, (unsigned int)__float_as_int(val));
}

// Pass 1 over edges: logits + per-dst running max
__global__ __launch_bounds__(256) void edge_logits_kernel(
    const int* __restrict__ src, const int* __restrict__ dst,
    const float* __restrict__ a_s, const float* __restrict__ a_d,
    float* __restrict__ logits, float* __restrict__ m, int E) {
  int e = blockIdx.x * blockDim.x + threadIdx.x;
  if (e >= E) return;
  int s = src[e], d = dst[e];
  float l = a_s[s] + a_d[d];
  l = (l > 0.0f) ? l : NEG_SLOPE * l;
  logits[e] = l;
  atomicMaxF(&m[d], l);
}

// Pass 2: exp(logit - max) + per-dst sum
__global__ __launch_bounds__(256) void edge_exp_kernel(
    const int* __restrict__ dst, const float* __restrict__ logits,
    const float* __restrict__ m, float* __restrict__ eexp,
    float* __restrict__ denom, int E) {
  int e = blockIdx.x * blockDim.x + threadIdx.x;
  if (e >= E) return;
  int d = dst[e];
  float v = __expf(logits[e] - m[d]);
  eexp[e] = v;
  atomicAdd(&denom[d], v);
}

// Pass 3: scatter messages. One wave32 per edge, float4 per lane (128 feats).
__global__ __launch_bounds__(256) void edge_message_kernel(
    const int* __restrict__ src, const int* __restrict__ dst,
    const float* __restrict__ eexp, const float* __restrict__ denom,
    const float* __restrict__ Hsrc, float* __restrict__ Hout, int E) {
  int e = blockIdx.x * 8 + (threadIdx.x >> 5);
  int lane = threadIdx.x & 31;
  if (e >= E) return;
  int s = src[e], d = dst[e];
  float alpha = eexp[e] / (denom[d] + 1e-16f);
  float4 hv = ((const float4*)(Hsrc + (size_t)s * HD))[lane];
  float* out = Hout + (size_t)d * HD + lane * 4;
  atomicAdd(out + 0, hv.x * alpha);
  atomicAdd(out + 1, hv.y * alpha);
  atomicAdd(out + 2, hv.z * alpha);
  atomicAdd(out + 3, hv.w * alpha);
}

// out = relu(out + bias)
__global__ void bias_relu_kernel(float* __restrict__ Hm,
                                 const float* __restrict__ bias, int total) {
  int i = blockIdx.x * blockDim.x + threadIdx.x;
  if (i >= total) return;
  float v = Hm[i] + bias[i & (HD - 1)];
  Hm[i] = v > 0.0f ? v : 0.0f;
}

// ---------------------------------------------------------------------------
// Global mean pool: scatter-sum per graph + count, then divide.
__global__ __launch_bounds__(256) void pool_sum_kernel(
    const float* __restrict__ Hin, const int* __restrict__ batch,
    float* __restrict__ pool, float* __restrict__ cnt, int N) {
  int n = blockIdx.x * 8 + (threadIdx.x >> 5);
  int lane = threadIdx.x & 31;
  if (n >= N) return;
  int g = batch[n];
  float4 hv = ((const float4*)(Hin + (size_t)n * HD))[lane];
  float* p = pool + (size_t)g * HD + lane * 4;
  atomicAdd(p + 0, hv.x);
  atomicAdd(p + 1, hv.y);
  atomicAdd(p + 2, hv.z);
  atomicAdd(p + 3, hv.w);
  if (lane == 0) atomicAdd(&cnt[g], 1.0f);
}

__global__ void pool_div_kernel(const float* __restrict__ pool,
                                const float* __restrict__ cnt,
                                float* __restrict__ out, int total) {
  int i = blockIdx.x * blockDim.x + threadIdx.x;
  if (i >= total) return;
  float c = cnt[i >> 7];
  out[i] = pool[i] / fmaxf(c, 1.0f);
}

// ---------------------------------------------------------------------------
extern "C" void kernel_launch(void* const* d_in, const int* in_sizes, int n_in,
                              void* d_out, int out_size, void* d_ws, size_t ws_size,
                              hipStream_t stream) {
  const float* x   = (const float*)d_in[0];
  const int* ei    = (const int*)d_in[1];
  const int* batch = (const int*)d_in[2];

  const int N = in_sizes[0] / D;        // 50000
  const int E = in_sizes[1] / 2;        // 800000
  const int G = out_size / HD;          // 512

  const int* src = ei;
  const int* dst = ei + E;

  // Workspace carve-up (floats). Total ~84 MB.
  float* ws = (float*)d_ws;
  size_t off = 0;
  float* bufA   = ws + off; off += (size_t)N * HD;
  float* bufB   = ws + off; off += (size_t)N * HD;
  float* h_src  = ws + off; off += (size_t)N * HD;
  float* a_s    = ws + off; off += N;
  float* a_d    = ws + off; off += N;
  float* m      = ws + off; off += N;
  float* denom  = ws + off; off += N;
  float* logits = ws + off; off += E;
  float* eexp   = ws + off; off += E;
  float* wcs    = ws + off; off += HD;
  float* wcd    = ws + off; off += HD;
  float* pool   = ws + off; off += (size_t)G * HD;
  float* cnt    = ws + off; off += G;

  const int nodeElems = N * HD;
  const int b256 = 256;

  const float* hin = x;
  for (int layer = 0; layer < 3; ++layer) {
    const float* Wsrc = (const float*)d_in[3 + 5 * layer + 0];
    const float* Wdst = (const float*)d_in[3 + 5 * layer + 1];
    const float* atts = (const float*)d_in[3 + 5 * layer + 2];
    const float* attd = (const float*)d_in[3 + 5 * layer + 3];
    const float* bias = (const float*)d_in[3 + 5 * layer + 4];
    float* hout = (layer == 0) ? bufA : (layer == 1 ? bufB : bufA);

    // Dense phase
    combine_att_kernel<<<1, 128, 0, stream>>>(Wsrc, atts, Wdst, attd, wcs, wcd);
    attvec_kernel<<<(N + 7) / 8, b256, 0, stream>>>(hin, wcs, wcd, a_s, a_d, N);
    gemm_hsrc_kernel<<<N / 16, b256, 0, stream>>>(hin, Wsrc, h_src, N);

    // Softmax state + output accumulator init
    init_node_kernel<<<(N + b256 - 1) / b256, b256, 0, stream>>>(m, denom, N);
    zero_kernel<<<(nodeElems + b256 - 1) / b256, b256, 0, stream>>>(hout, nodeElems);

    // Edge phase
    edge_logits_kernel<<<(E + b256 - 1) / b256, b256, 0, stream>>>(src, dst, a_s, a_d, logits, m, E);
    edge_exp_kernel<<<(E + b256 - 1) / b256, b256, 0, stream>>>(dst, logits, m, eexp, denom, E);
    edge_message_kernel<<<(E + 7) / 8, b256, 0, stream>>>(src, dst, eexp, denom, h_src, hout, E);

    bias_relu_kernel<<<(nodeElems + b256 - 1) / b256, b256, 0, stream>>>(hout, bias, nodeElems);
    hin = hout;
  }

  // Global mean pool
  zero_kernel<<<((G * HD) + b256 - 1) / b256, b256, 0, stream>>>(pool, G * HD);
  zero_kernel<<<(G + b256 - 1) / b256, b256, 0, stream>>>(cnt, G);
  pool_sum_kernel<<<(N + 7) / 8, b256, 0, stream>>>(hin, batch, pool, cnt, N);
  pool_div_kernel<<<((G * HD) + b256 - 1) / b256, b256, 0, stream>>>(pool, cnt, (float*)d_out, G * HD);
}